// CLRNetIoULoss_68796786147813
// MI455X (gfx1250) — compile-verified
//
#include <hip/hip_runtime.h>
#include <stdint.h>

// ---------------------------------------------------------------------------
// CLRNet IoU loss, MI455X (gfx1250).
// Memory-bound: 576 MB read @ 23.3 TB/s => ~25us floor. Strategy: async
// global->LDS streaming (CDNA5 ASYNCcnt path) of contiguous b128 chunks,
// non-temporal (dataset is 3x L2 and read exactly once), padded-row LDS
// layout (stride 76 floats -> conflict-free b128 LDS reads per 16-lane
// phase), one thread reduces one full 72-element row.
// ---------------------------------------------------------------------------

#define NR_COLS       72
#define ROWS_PER_BLK  128
#define LDS_STRIDE    76          // floats; 72 padded to 76
#define F4_PER_ROW    (NR_COLS / 4)   // 18
#define W_HALF        0.01875f    // 15/800

// Issue one async 16-byte global->LDS copy, non-temporal (streamed once).
// lds_byte  : workgroup-relative LDS byte address (per-lane VGPR)
// base      : uniform 64-bit global base pointer (SGPR pair)
// glb_byte  : per-lane 32-bit unsigned byte offset (VGPR)
__device__ __forceinline__ void async_ld_b128(uint32_t lds_byte,
                                              const float* base,
                                              uint32_t glb_byte) {
    asm volatile("global_load_async_to_lds_b128 %0, %1, %2 th:TH_LOAD_NT"
                 :
                 : "v"(lds_byte), "v"(glb_byte), "s"(base)
                 : "memory");
}

__device__ __forceinline__ void wait_async_zero() {
    asm volatile("s_wait_asynccnt 0" ::: "memory");
}

// Generic pointer to LDS: low 32 bits are the workgroup-relative byte offset.
__device__ __forceinline__ uint32_t lds_byte_off(const void* p) {
    return (uint32_t)(uintptr_t)p;
}

extern "C" __global__ void __launch_bounds__(ROWS_PER_BLK)
iou_partial_kernel(const float* __restrict__ pred,
                   const float* __restrict__ targ,
                   float* __restrict__ partial,
                   int n_rows) {
    __shared__ float shP[ROWS_PER_BLK * LDS_STRIDE];
    __shared__ float shT[ROWS_PER_BLK * LDS_STRIDE];
    __shared__ float red[ROWS_PER_BLK];

    const int t    = threadIdx.x;
    const int row0 = blockIdx.x * ROWS_PER_BLK;
    const int rows_here = min(ROWS_PER_BLK, n_rows - row0);

    const uint32_t shP0 = lds_byte_off(shP);
    const uint32_t shT0 = lds_byte_off(shT);
    const uint32_t gbase = (uint32_t)row0 * (uint32_t)(NR_COLS * 4);

    // ---- async stage: contiguous global float4 stream -> padded LDS rows ----
    const int f_limit = rows_here * F4_PER_ROW;          // valid float4s in tile
    for (int f = t; f < f_limit; f += ROWS_PER_BLK) {
        int r = f / F4_PER_ROW;                          // row within tile
        int q = f - r * F4_PER_ROW;                      // float4 chunk in row
        uint32_t ldsb = (uint32_t)(r * (LDS_STRIDE * 4) + q * 16);
        uint32_t gb   = gbase + (uint32_t)f * 16u;
        async_ld_b128(shP0 + ldsb, pred, gb);
        async_ld_b128(shT0 + ldsb, targ, gb);
    }
    wait_async_zero();
    __syncthreads();

    // ---- per-thread full-row reduction out of LDS ----
    float contrib = 0.0f;
    if (t < rows_here) {
        const float4* rp4 = (const float4*)(shP + t * LDS_STRIDE);
        const float4* rt4 = (const float4*)(shT + t * LDS_STRIDE);
        float so = 0.0f, su = 0.0f;
        int   tp = 0,    er = 0;

        #pragma unroll 6
        for (int jq = 0; jq < F4_PER_ROW; ++jq) {
            float4 P = rp4[jq];
            float4 Q = rt4[jq];
            #pragma unroll
            for (int c = 0; c < 4; ++c) {
                float p = (c == 0) ? P.x : (c == 1) ? P.y : (c == 2) ? P.z : P.w;
                float q = (c == 0) ? Q.x : (c == 1) ? Q.y : (c == 2) ? Q.z : Q.w;
                bool vp = (p >= 0.0f) & (p < 1.0f);
                bool vt = (q >= 0.0f) & (q < 1.0f);
                float px1 = p - W_HALF, px2 = p + W_HALF;
                float tx1 = q - W_HALF, tx2 = q + W_HALF;
                float ovr = fminf(px2, tx2) - fmaxf(px1, tx1);
                float uni = fmaxf(px2, tx2) - fminf(px1, tx1);
                bool both = vp & vt;
                so += both ? ovr : 0.0f;
                su += both ? uni : 0.0f;
                tp += (int)both;
                er += (int)(vp != vt);
            }
        }
        float iou = so / (su + 1e-9f);
        float tpf = (float)tp, erf = (float)er;
        if ((tpf > erf) && (er > 0))
            iou *= (1.0f - erf / (tpf + 1e-9f));
        contrib = 1.0f - iou;
    }

    // ---- deterministic block tree reduction ----
    red[t] = contrib;
    __syncthreads();
    #pragma unroll
    for (int s = ROWS_PER_BLK / 2; s > 0; s >>= 1) {
        if (t < s) red[t] += red[t + s];
        __syncthreads();
    }
    if (t == 0) partial[blockIdx.x] = red[0];
}

extern "C" __global__ void __launch_bounds__(256)
iou_final_kernel(const float* __restrict__ partial, int n,
                 float inv_n_rows, float* __restrict__ out) {
    __shared__ float red[256];
    float s = 0.0f;
    for (int i = threadIdx.x; i < n; i += 256) s += partial[i];
    red[threadIdx.x] = s;
    __syncthreads();
    #pragma unroll
    for (int k = 128; k > 0; k >>= 1) {
        if (threadIdx.x < k) red[threadIdx.x] += red[threadIdx.x + k];
        __syncthreads();
    }
    if (threadIdx.x == 0) out[0] = red[0] * inv_n_rows;   // LOSS_WEIGHT == 1
}

extern "C" void kernel_launch(void* const* d_in, const int* in_sizes, int n_in,
                              void* d_out, int out_size, void* d_ws, size_t ws_size,
                              hipStream_t stream) {
    const float* pred = (const float*)d_in[0];
    const float* targ = (const float*)d_in[1];
    const int n_rows  = in_sizes[0] / NR_COLS;            // 1,000,000
    const int nblk    = (n_rows + ROWS_PER_BLK - 1) / ROWS_PER_BLK;
    float* partial    = (float*)d_ws;                     // nblk floats (~31 KB)

    iou_partial_kernel<<<nblk, ROWS_PER_BLK, 0, stream>>>(pred, targ, partial, n_rows);
    iou_final_kernel<<<1, 256, 0, stream>>>(partial, nblk,
                                            1.0f / (float)n_rows, (float*)d_out);
}